// Net_33586644254827
// MI455X (gfx1250) — compile-verified
//
#include <hip/hip_runtime.h>

// ---------------- problem constants ----------------
#define NN    50000
#define EE    800000
#define GG    128
#define F_IN  310
#define HH    128
#define NHID  64
#define CC    2
#define EP    (EE + NN)          // edges + self loops

#define SELU_SCALE 1.0507009873554805f
#define SELU_ALPHA 1.6732632423543772f
#define NEG_SLOPE  0.2f

// GEMM tiling: 5 waves/block (3125 row-tiles = 625 blocks x 5, all waves active)
#define GEMM_WAVES   5
#define GEMM_THREADS (32 * GEMM_WAVES)
#define KPANEL       32
// LDS holds B pairs: element (q, c) = (B[2q][c], B[2q+1][c]) as float2.
// Pair-row stride 144 float2: 144 ≡ 16 (mod 32) -> the two half-waves
// (q and q+1) land on disjoint 32-bank halves -> conflict-free ds_load_b64.
#define S2 144

typedef float v2f __attribute__((ext_vector_type(2)));
typedef float v8f __attribute__((ext_vector_type(8)));

__device__ __forceinline__ float selu_f(float x) {
    return SELU_SCALE * (x > 0.0f ? x : SELU_ALPHA * (__expf(x) - 1.0f));
}
// order-preserving float<->uint map for atomicMax on floats
__device__ __forceinline__ unsigned f2ord(float x) {
    unsigned u = __float_as_uint(x);
    return (u & 0x80000000u) ? ~u : (u | 0x80000000u);
}
__device__ __forceinline__ float ord2f(unsigned e) {
    return (e & 0x80000000u) ? __uint_as_float(e & 0x7fffffffu)
                             : __uint_as_float(~e);
}
__device__ __forceinline__ void edge_sd(const int* __restrict__ ei, int i,
                                        int& s, int& d) {
    if (i < EE) { s = ei[i]; d = ei[EE + i]; }
    else        { s = i - EE; d = i - EE; }   // self loops
}

// ---------------------------------------------------------------------------
// GEMM: C[nRows x 128] = A[nRows x K] * B[K x 128], fp32 WMMA 16x16x4.
// One wave per 16-row stripe; B staged in LDS per block in pair-interleaved
// layout. Main loop: full panels only (zero guards). Tail panel peeled.
// nRows must equal 16*GEMM_WAVES*gridDim (50000 = 625*80).
// ---------------------------------------------------------------------------
__global__ __launch_bounds__(GEMM_THREADS)
void gemm_wmma_f32(const float* __restrict__ A, int K,
                   const float* __restrict__ B,
                   float* __restrict__ Cout) {
    __shared__ float ldsB[(KPANEL / 2) * S2 * 2];   // 16*144 float2 = 18 KB

    const int lane = threadIdx.x & 31;
    const int wave = threadIdx.x >> 5;
    const int l16  = lane & 15;
    const int hi   = lane >> 4;                   // 0: lanes 0-15, 1: lanes 16-31
    const int m0   = (blockIdx.x * GEMM_WAVES + wave) * 16;

    v8f acc[8];
#pragma unroll
    for (int t = 0; t < 8; ++t) acc[t] = (v8f){0,0,0,0,0,0,0,0};

    const float* Arow = A + (size_t)(m0 + l16) * (size_t)K;

    const int numFull = K / KPANEL;
    for (int p = 0; p < numFull; ++p) {
        const int k0p = p * KPANEL;

        // ---- stage full panel: B[k0p..k0p+31][:] -> LDS pair layout (no guards)
        for (int i = threadIdx.x; i < KPANEL * (HH / 4); i += GEMM_THREADS) {
            const int k  = i >> 5;              // 32 float4 per B row
            const int c4 = (i & 31) * 4;
            const float4 v = ((const float4*)(B + (size_t)(k0p + k) * HH))[i & 31];
            float* dst = ldsB + ((k >> 1) * S2 + c4) * 2 + (k & 1);
            dst[0] = v.x; dst[2] = v.y; dst[4] = v.z; dst[6] = v.w;
        }
        __syncthreads();

        // ---- single unguarded WMMA chain (in-place accumulation) ----
#pragma unroll
        for (int kk8 = 0; kk8 < KPANEL / 4; ++kk8) {
            const int ka = kk8 * 4 + 2 * hi;    // local k of this lane's fragment
            const int q  = kk8 * 2 + hi;        // pair-row in LDS
            v2f a;
            a.x = Arow[k0p + ka];
            a.y = Arow[k0p + ka + 1];
#pragma unroll
            for (int t = 0; t < 8; ++t) {
                const v2f b = *(const v2f*)(ldsB + (q * S2 + t * 16 + l16) * 2);
                acc[t] = __builtin_amdgcn_wmma_f32_16x16x4_f32(
                    false, a, false, b, (short)0, acc[t], false, false);
            }
        }
        __syncthreads();
    }

    // ---- peeled tail panel (runs at most once) ----
    if (numFull * KPANEL < K) {
        const int k0p = numFull * KPANEL;
        for (int i = threadIdx.x; i < KPANEL * (HH / 4); i += GEMM_THREADS) {
            const int k  = i >> 5;
            const int c4 = (i & 31) * 4;
            const int kg = k0p + k;
            float4 v = make_float4(0.f, 0.f, 0.f, 0.f);
            if (kg < K) v = ((const float4*)(B + (size_t)kg * HH))[i & 31];
            float* dst = ldsB + ((k >> 1) * S2 + c4) * 2 + (k & 1);
            dst[0] = v.x; dst[2] = v.y; dst[4] = v.z; dst[6] = v.w;
        }
        __syncthreads();
#pragma unroll
        for (int kk8 = 0; kk8 < KPANEL / 4; ++kk8) {
            const int ka = kk8 * 4 + 2 * hi;
            const int q  = kk8 * 2 + hi;
            const int kg = k0p + ka;
            v2f a;
            a.x = (kg     < K) ? Arow[kg]     : 0.0f;
            a.y = (kg + 1 < K) ? Arow[kg + 1] : 0.0f;
#pragma unroll
            for (int t = 0; t < 8; ++t) {
                const v2f b = *(const v2f*)(ldsB + (q * S2 + t * 16 + l16) * 2);
                acc[t] = __builtin_amdgcn_wmma_f32_16x16x4_f32(
                    false, a, false, b, (short)0, acc[t], false, false);
            }
        }
    }

    // C layout: VGPR r, lanes 0-15 -> M=r, lanes 16-31 -> M=8+r; N = lane%16
#pragma unroll
    for (int t = 0; t < 8; ++t) {
#pragma unroll
        for (int r = 0; r < 8; ++r) {
            const int rr = m0 + r + 8 * hi;
            Cout[(size_t)rr * HH + t * 16 + l16] = acc[t][r];
        }
    }
}

// ---------------------------------------------------------------------------
// Per-node attention dots: outa[n] = h[n,:].a ; outb[n] = h[n,:].b  (wave/node)
// ---------------------------------------------------------------------------
__global__ __launch_bounds__(256)
void node_dots(const float* __restrict__ h, const float* __restrict__ va,
               const float* __restrict__ vb, float* __restrict__ outa,
               float* __restrict__ outb) {
    const int node = (blockIdx.x * blockDim.x + threadIdx.x) >> 5;
    const int lane = threadIdx.x & 31;
    if (node >= NN) return;
    const float* row = h + (size_t)node * HH;
    float sa = 0.0f, sb = 0.0f;
#pragma unroll
    for (int j = 0; j < 4; ++j) {
        const int c = lane + 32 * j;
        const float x = row[c];
        sa += x * va[c];
        sb += x * vb[c];
    }
#pragma unroll
    for (int off = 16; off >= 1; off >>= 1) {
        sa += __shfl_xor(sa, off, 32);
        sb += __shfl_xor(sb, off, 32);
    }
    if (lane == 0) { outa[node] = sa; outb[node] = sb; }
}

// ---------------------------------------------------------------------------
// Edge pass 1: e = leaky_relu(asrc[s]+adst[d]); segment max via ordered atomicMax
// ---------------------------------------------------------------------------
__global__ __launch_bounds__(256)
void edge_max_k(const int* __restrict__ ei, const float* __restrict__ asrc,
                const float* __restrict__ adst, float* __restrict__ ebuf,
                unsigned* __restrict__ m_u) {
    const int i = blockIdx.x * blockDim.x + threadIdx.x;
    if (i >= EP) return;
    int s, d; edge_sd(ei, i, s, d);
    float e = asrc[s] + adst[d];
    e = e > 0.0f ? e : NEG_SLOPE * e;
    ebuf[i] = e;
    atomicMax(&m_u[d], f2ord(e));
}

// Edge pass 2: ex = exp(e - m[d]); segment sum
__global__ __launch_bounds__(256)
void edge_exp_k(const int* __restrict__ ei, float* __restrict__ ebuf,
                const unsigned* __restrict__ m_u, float* __restrict__ ssum) {
    const int i = blockIdx.x * blockDim.x + threadIdx.x;
    if (i >= EP) return;
    int s, d; edge_sd(ei, i, s, d);
    (void)s;
    const float ex = __expf(ebuf[i] - ord2f(m_u[d]));
    ebuf[i] = ex;
    atomicAdd(&ssum[d], ex);
}

// Edge pass 3: agg[d,:] += h[s,:] * (ex / sum[d]) ; wave per edge, float4 gather
__global__ __launch_bounds__(256)
void edge_agg_k(const int* __restrict__ ei, const float* __restrict__ ebuf,
                const float* __restrict__ ssum, const float* __restrict__ h,
                float* __restrict__ agg) {
    const int i    = blockIdx.x * 8 + (threadIdx.x >> 5);
    const int lane = threadIdx.x & 31;
    if (i >= EP) return;
    int s, d; edge_sd(ei, i, s, d);
    const float w = ebuf[i] / ssum[d];
    const float4 hv = ((const float4*)(h + (size_t)s * HH))[lane];
    float* out = agg + (size_t)d * HH + lane * 4;
    atomicAdd(out + 0, hv.x * w);
    atomicAdd(out + 1, hv.y * w);
    atomicAdd(out + 2, hv.z * w);
    atomicAdd(out + 3, hv.w * w);
}

// x = selu(x + bias[col])   (in place over N x 128)
__global__ __launch_bounds__(256)
void bias_selu_k(float* __restrict__ buf, const float* __restrict__ bias) {
    const int i = blockIdx.x * blockDim.x + threadIdx.x;
    if (i >= NN * HH) return;
    buf[i] = selu_f(buf[i] + bias[i & (HH - 1)]);
}

// Pool: pooled[batch[n],:] += x[n,:]; cnt[batch[n]] += 1   (wave per node)
__global__ __launch_bounds__(256)
void pool_k(const float* __restrict__ x, const int* __restrict__ batch,
            float* __restrict__ pooled, float* __restrict__ cnt) {
    const int node = (blockIdx.x * blockDim.x + threadIdx.x) >> 5;
    const int lane = threadIdx.x & 31;
    if (node >= NN) return;
    const int g = batch[node];
    const float4 hv = ((const float4*)(x + (size_t)node * HH))[lane];
    float* out = pooled + (size_t)g * HH + lane * 4;
    atomicAdd(out + 0, hv.x);
    atomicAdd(out + 1, hv.y);
    atomicAdd(out + 2, hv.z);
    atomicAdd(out + 3, hv.w);
    if (lane == 0) atomicAdd(&cnt[g], 1.0f);
}

// z[g,j] = selu( fc1_b[j] + sum_k selu(pooled[g,k]/max(cnt,1)) * fc1_w[k,j] )
__global__ __launch_bounds__(256)
void fc1_k(const float* __restrict__ pooled, const float* __restrict__ cnt,
           const float* __restrict__ w, const float* __restrict__ b,
           float* __restrict__ z) {
    const int tid = blockIdx.x * blockDim.x + threadIdx.x;
    if (tid >= GG * NHID) return;
    const int g = tid / NHID, j = tid % NHID;
    const float inv = 1.0f / fmaxf(cnt[g], 1.0f);
    float acc = b[j];
    for (int k = 0; k < HH; ++k)
        acc += selu_f(pooled[g * HH + k] * inv) * w[k * NHID + j];
    z[g * NHID + j] = selu_f(acc);
}

// logits + log_softmax (C = 2)
__global__ __launch_bounds__(128)
void head_k(const float* __restrict__ z, const float* __restrict__ w,
            const float* __restrict__ b, float* __restrict__ out) {
    const int g = blockIdx.x * blockDim.x + threadIdx.x;
    if (g >= GG) return;
    float l0 = b[0], l1 = b[1];
    for (int j = 0; j < NHID; ++j) {
        const float zj = z[g * NHID + j];
        l0 += zj * w[j * CC + 0];
        l1 += zj * w[j * CC + 1];
    }
    const float m   = fmaxf(l0, l1);
    const float lse = m + __logf(__expf(l0 - m) + __expf(l1 - m));
    out[g * CC + 0] = l0 - lse;
    out[g * CC + 1] = l1 - lse;
}

// ---------------------------------------------------------------------------
extern "C" void kernel_launch(void* const* d_in, const int* in_sizes, int n_in,
                              void* d_out, int out_size, void* d_ws, size_t ws_size,
                              hipStream_t stream) {
    (void)in_sizes; (void)n_in; (void)out_size; (void)ws_size;
    const float* x      = (const float*)d_in[0];
    const int*   ei     = (const int*)  d_in[1];
    const int*   batch  = (const int*)  d_in[2];
    const float* W1     = (const float*)d_in[3];
    const float* asrc1v = (const float*)d_in[4];
    const float* adst1v = (const float*)d_in[5];
    const float* b1     = (const float*)d_in[6];
    const float* W2     = (const float*)d_in[7];
    const float* asrc2v = (const float*)d_in[8];
    const float* adst2v = (const float*)d_in[9];
    const float* b2     = (const float*)d_in[10];
    const float* fc1w   = (const float*)d_in[11];
    const float* fc1b   = (const float*)d_in[12];
    const float* fc2w   = (const float*)d_in[13];
    const float* fc2b   = (const float*)d_in[14];
    float* out = (float*)d_out;

    // workspace layout (floats)
    float* ws = (float*)d_ws;
    const size_t NHf = (size_t)NN * HH;           // 6.4M
    float*    buf0   = ws;                        // h of current layer
    float*    buf1   = ws + NHf;                  // aggregation / activations
    float*    asrc   = ws + 2 * NHf;
    float*    adst   = asrc + NN;
    unsigned* m_u    = (unsigned*)(adst + NN);
    float*    ssum   = (float*)m_u + NN;
    float*    ebuf   = ssum + NN;                 // EP floats
    float*    pooled = ebuf + EP;                 // G*H
    float*    cnt    = pooled + (size_t)GG * HH;  // G
    float*    zbuf   = cnt + GG;                  // G*NHID

    const int gemmGrid = (NN / 16) / GEMM_WAVES;      // 625, all waves active
    const int waveGrid = (NN * 32 + 255) / 256;       // wave-per-node kernels
    const int edgeGrid = (EP + 255) / 256;
    const int aggGrid  = (EP + 7) / 8;
    const int nhGrid   = (NN * HH + 255) / 256;

    // ---- init (graph-capture-safe async memsets) ----
    hipMemsetAsync(buf1,   0, NHf * sizeof(float), stream);
    hipMemsetAsync(m_u,    0, NN * sizeof(unsigned), stream);
    hipMemsetAsync(ssum,   0, NN * sizeof(float), stream);
    hipMemsetAsync(pooled, 0, (size_t)GG * HH * sizeof(float), stream);
    hipMemsetAsync(cnt,    0, GG * sizeof(float), stream);

    // ---- layer 1 ----
    gemm_wmma_f32<<<gemmGrid, GEMM_THREADS, 0, stream>>>(x, F_IN, W1, buf0);
    node_dots<<<waveGrid, 256, 0, stream>>>(buf0, asrc1v, adst1v, asrc, adst);
    edge_max_k<<<edgeGrid, 256, 0, stream>>>(ei, asrc, adst, ebuf, m_u);
    edge_exp_k<<<edgeGrid, 256, 0, stream>>>(ei, ebuf, m_u, ssum);
    edge_agg_k<<<aggGrid, 256, 0, stream>>>(ei, ebuf, ssum, buf0, buf1);
    bias_selu_k<<<nhGrid, 256, 0, stream>>>(buf1, b1);   // buf1 = selu(agg + b1)

    // ---- layer 2 ----
    gemm_wmma_f32<<<gemmGrid, GEMM_THREADS, 0, stream>>>(buf1, HH, W2, buf0);
    hipMemsetAsync(m_u,  0, NN * sizeof(unsigned), stream);
    hipMemsetAsync(ssum, 0, NN * sizeof(float), stream);
    hipMemsetAsync(buf1, 0, NHf * sizeof(float), stream);
    node_dots<<<waveGrid, 256, 0, stream>>>(buf0, asrc2v, adst2v, asrc, adst);
    edge_max_k<<<edgeGrid, 256, 0, stream>>>(ei, asrc, adst, ebuf, m_u);
    edge_exp_k<<<edgeGrid, 256, 0, stream>>>(ei, ebuf, m_u, ssum);
    edge_agg_k<<<aggGrid, 256, 0, stream>>>(ei, ebuf, ssum, buf0, buf1);
    bias_selu_k<<<nhGrid, 256, 0, stream>>>(buf1, b2);   // buf1 = selu(agg + b2)

    // ---- pool + head ----
    pool_k<<<waveGrid, 256, 0, stream>>>(buf1, batch, pooled, cnt);
    fc1_k<<<(GG * NHID + 255) / 256, 256, 0, stream>>>(pooled, cnt, fc1w, fc1b, zbuf);
    head_k<<<1, 128, 0, stream>>>(zbuf, fc2w, fc2b, out);
}